// emd_standard_decomp_59691455479868
// MI455X (gfx1250) — compile-verified
//
#include <hip/hip_runtime.h>
#include <math.h>

#define TWO_PI 6.28318530717958647692f

// ---- problem constants (B=32, L=4096, D=64 -> N = 2^23) ----
#define NTOT   (1 << 23)
#define LDIM   4096
#define DDIM   64

typedef __attribute__((ext_vector_type(2))) float v2f;
typedef __attribute__((ext_vector_type(8))) float v8f;

__device__ __forceinline__ float2 cmul(float2 a, float2 b) {
    return make_float2(a.x * b.x - a.y * b.y, a.x * b.y + a.y * b.x);
}

// ---------------------------------------------------------------------------
// 1) Forward Stockham radix-8 first stage (Ns=1, no twiddles), fused with the
//    rectification: input is the real signal, xr[q] = max(sgn*x,0), imag = 0.
//    dir = -1 (forward); DFT-8 angles are multiples of 45 deg -> const table.
// ---------------------------------------------------------------------------
__global__ void __launch_bounds__(256)
k_fft_r8_fwd(const float* __restrict__ x, float2* __restrict__ dst, float sgn) {
    const int NdR = NTOT >> 3;
    int t = blockIdx.x * 256 + threadIdx.x;       // t in [0, N/8)
    float xr[8];
#pragma unroll
    for (int q = 0; q < 8; ++q) {
        float v = sgn * x[t + q * NdR];
        xr[q] = v > 0.0f ? v : 0.0f;
    }
    const float R = 0.70710678118654752440f;      // cis(-2*pi*m/8)
    const float wc[8] = { 1.0f,  R, 0.0f, -R, -1.0f, -R,  0.0f,  R };
    const float ws[8] = { 0.0f, -R, -1.0f, -R, 0.0f,  R,  1.0f,  R };
    long obase = (long)t << 3;                    // idxD = 8*t + p
#pragma unroll
    for (int p = 0; p < 8; ++p) {
        float ar = 0.0f, ai = 0.0f;
#pragma unroll
        for (int q = 0; q < 8; ++q) {
            int m = (p * q) & 7;
            ar += xr[q] * wc[m];
            ai += xr[q] * ws[m];
        }
        dst[obase + p] = make_float2(ar, ai);
    }
}

// ---------------------------------------------------------------------------
// 2) Inverse Stockham radix-8 first stage (Ns=1) with the Hilbert spectral
//    mask h(k) (1 at k=0,N/2; 2 for 0<k<N/2; 0 above) fused into the loads.
//    dir = +1 (inverse); 1/N scaling is folded into the magnitude pass.
// ---------------------------------------------------------------------------
__global__ void __launch_bounds__(256)
k_fft_r8_inv(const float2* __restrict__ src, float2* __restrict__ dst) {
    const int NdR = NTOT >> 3;
    int t = blockIdx.x * 256 + threadIdx.x;
    float2 xc[8];
#pragma unroll
    for (int q = 0; q < 8; ++q) {
        int idx = t + q * NdR;
        float2 v = src[idx];
        float h = (idx == 0 || idx == (NTOT >> 1)) ? 1.0f
                  : (idx < (NTOT >> 1) ? 2.0f : 0.0f);
        v.x *= h; v.y *= h;
        xc[q] = v;
    }
    const float R = 0.70710678118654752440f;      // cis(+2*pi*m/8)
    const float wc[8] = { 1.0f,  R,  0.0f, -R, -1.0f, -R,  0.0f,  R };
    const float ws[8] = { 0.0f,  R,  1.0f,  R,  0.0f, -R, -1.0f, -R };
    long obase = (long)t << 3;
#pragma unroll
    for (int p = 0; p < 8; ++p) {
        float ar = 0.0f, ai = 0.0f;
#pragma unroll
        for (int q = 0; q < 8; ++q) {
            int m = (p * q) & 7;
            ar += xc[q].x * wc[m] - xc[q].y * ws[m];
            ai += xc[q].x * ws[m] + xc[q].y * wc[m];
        }
        dst[obase + p] = make_float2(ar, ai);
    }
}

// ---------------------------------------------------------------------------
// 3) Stockham radix-16 stage via V_WMMA_F32_16X16X4_F32 (tcFFT style).
//    One wave = 16 butterflies:  Y(16x16) = W(16x16) x Xtw(16x16) complex,
//    K=16 split into 4 chained K=4 WMMAs; complex -> 4 real products
//    => 16 v_wmma per wave iteration.
//    f32 A/B fragment: lane l holds (row/col = l&15, K = 2*(l>>4)+v), v=0,1.
//    f32 C/D fragment: lane l, reg r holds (M = r + 8*(l>>4), N = l&15).
//    W row entries and twiddles are generated as powers of one cis() each:
//      W[c][q] = u1^q, u1 = cis(dir*2pi*c/16)
//      tw(q)   = w1^q, w1 = cis(dir*2pi*jm/(16*Ns))
//    so only 2 sincos per lane; the rest is dual-issuable FMA cmuls.
//    Exact grid, no divergence: EXEC all 1s as WMMA requires.
// ---------------------------------------------------------------------------
__global__ void __launch_bounds__(256)
k_fft_r16_wmma(const float2* __restrict__ src, float2* __restrict__ dst,
               int nsShift, float dir, float inv16Ns) {
    const int NdR = NTOT >> 4;                    // N/16
    const int Ns  = 1 << nsShift;
    int lane = threadIdx.x & 31;
    int wave = blockIdx.x * 8 + (threadIdx.x >> 5);
    int t0   = wave << 4;
    int c    = lane & 15;                         // butterfly column / W row
    int half = lane >> 4;                         // 0 or 1
    int j    = t0 + c;
    int jm   = j & (Ns - 1);
    int k    = j >> nsShift;

    // generators
    float su, cu; __sincosf(dir * (TWO_PI / 16.0f) * (float)c, &su, &cu);
    float sw, cw; __sincosf(dir * TWO_PI * ((float)jm * inv16Ns), &sw, &cw);
    float2 one = make_float2(1.0f, 0.0f);
    float2 u1 = make_float2(cu, su), w1 = make_float2(cw, sw);
    float2 u2 = cmul(u1, u1),        w2 = cmul(w1, w1);
    float2 u4 = cmul(u2, u2),        w4 = cmul(w2, w2);
    float2 ue0 = half ? u2 : one,    we0 = half ? w2 : one;   // q = 2*half + v
    float2 ue1 = cmul(ue0, u1),      we1 = cmul(we0, w1);
    float2 uf = one, wf = one;                                // (u4)^cc, (w4)^cc

    v8f yr = (v8f)0.0f;
    v8f yi = (v8f)0.0f;

#pragma unroll
    for (int cc = 0; cc < 4; ++cc) {
        int q0 = 4 * cc + 2 * half;
        float2 a0 = cmul(uf, ue0), a1 = cmul(uf, ue1);        // W[c][q0], W[c][q0+1]
        float2 t0w = cmul(wf, we0), t1w = cmul(wf, we1);      // tw(q0), tw(q0+1)
        float2 x0 = src[j + q0 * NdR];
        float2 x1 = src[j + (q0 + 1) * NdR];
        float2 b0 = cmul(t0w, x0), b1 = cmul(t1w, x1);        // twiddled data
        v2f ar, ai, br, bi, nbi;
        ar[0] = a0.x; ar[1] = a1.x;  ai[0] = a0.y; ai[1] = a1.y;
        br[0] = b0.x; br[1] = b1.x;  bi[0] = b0.y; bi[1] = b1.y;
        nbi[0] = -b0.y; nbi[1] = -b1.y;
        yr = __builtin_amdgcn_wmma_f32_16x16x4_f32(false, ar, false, br,
                                                   (short)0, yr, false, false);
        yr = __builtin_amdgcn_wmma_f32_16x16x4_f32(false, ai, false, nbi,
                                                   (short)0, yr, false, false);
        yi = __builtin_amdgcn_wmma_f32_16x16x4_f32(false, ar, false, bi,
                                                   (short)0, yi, false, false);
        yi = __builtin_amdgcn_wmma_f32_16x16x4_f32(false, ai, false, br,
                                                   (short)0, yi, false, false);
        if (cc < 3) { uf = cmul(uf, u4); wf = cmul(wf, w4); }
    }

    // scatter: dst[k*16*Ns + p*Ns + jm], p = r + 8*half, column j = t0+(l&15)
    long base = ((long)k << (nsShift + 4)) + jm;
#pragma unroll
    for (int r = 0; r < 8; ++r) {
        int p = r + 8 * half;
        float2 o; o.x = yr[r]; o.y = yi[r];
        dst[base + ((long)p << nsShift)] = o;
    }
}

// ---------------------------------------------------------------------------
// 4) envelope magnitude + combine (1/N inverse-FFT scaling folded in).
//    mode 0: out2[i] =  |A[i]|/N                    (env_max, up signal)
//    mode 1: out2[i] = (out2[i] - |A[i]|/N) * 0.5   (avg envelope)
// ---------------------------------------------------------------------------
__global__ void __launch_bounds__(256)
k_env(const float2* __restrict__ A, float* __restrict__ out2, int mode) {
    int i = blockIdx.x * 256 + threadIdx.x;
    float2 a = A[i];
    float m = sqrtf(a.x * a.x + a.y * a.y) * (1.0f / (float)NTOT);
    out2[i] = (mode == 0) ? m : (out2[i] - m) * 0.5f;
}

// ---------------------------------------------------------------------------
// 5) per-(b,d) std over L (ddof=1) of imf = x - avg;  stds[bd] = std + 0.01
// ---------------------------------------------------------------------------
__global__ void __launch_bounds__(256)
k_std(const float* __restrict__ x, const float* __restrict__ avg,
      float* __restrict__ stds) {
    __shared__ float s_sum[256];
    __shared__ float s_sq[256];
    int bd = blockIdx.x;
    int b = bd >> 6, d = bd & 63;
    long base = (long)b * LDIM * DDIM + d;
    float sum = 0.0f, sq = 0.0f;
    for (int l = threadIdx.x; l < LDIM; l += 256) {
        long idx = base + (long)l * DDIM;
        float v = x[idx] - avg[idx];
        sum += v; sq += v * v;
    }
    s_sum[threadIdx.x] = sum; s_sq[threadIdx.x] = sq;
    __syncthreads();
    for (int off = 128; off > 0; off >>= 1) {
        if (threadIdx.x < off) {
            s_sum[threadIdx.x] += s_sum[threadIdx.x + off];
            s_sq[threadIdx.x]  += s_sq[threadIdx.x + off];
        }
        __syncthreads();
    }
    if (threadIdx.x == 0) {
        float mean = s_sum[0] * (1.0f / (float)LDIM);
        float var  = (s_sq[0] - s_sum[0] * mean) * (1.0f / (float)(LDIM - 1));
        stds[bd] = sqrtf(fmaxf(var, 0.0f)) + 0.01f;
    }
}

// ---------------------------------------------------------------------------
// 6) out1[i] = (x[i] - avg[i]) / stds[bd]
// ---------------------------------------------------------------------------
__global__ void __launch_bounds__(256)
k_final(const float* __restrict__ x, const float* __restrict__ avg,
        const float* __restrict__ stds, float* __restrict__ out1) {
    int i = blockIdx.x * 256 + threadIdx.x;
    int b = i >> 18;                // L*D = 2^18
    int d = i & 63;
    out1[i] = (x[i] - avg[i]) / stds[(b << 6) | d];
}

// ---------------------------------------------------------------------------
extern "C" void kernel_launch(void* const* d_in, const int* in_sizes, int n_in,
                              void* d_out, int out_size, void* d_ws, size_t ws_size,
                              hipStream_t stream) {
    (void)in_sizes; (void)n_in; (void)out_size; (void)ws_size;
    const float* x = (const float*)d_in[0];
    float* out1 = (float*)d_out;            // imf / std
    float* out2 = out1 + (size_t)NTOT;      // avg_envelope

    float2* A    = (float2*)d_ws;                   // 64 MiB
    float2* Bb   = A + (size_t)NTOT;                // 64 MiB
    float*  stds = (float*)(Bb + (size_t)NTOT);     // 8 KiB

    dim3 blk(256);
    const int gElem = NTOT / 256;           // 32768 blocks, elementwise
    const int gR8   = (NTOT / 8) / 256;     // 4096 blocks
    const int gR16  = (NTOT / 16 / 16) / 8; // 4096 blocks (8 waves/block)

    for (int pass = 0; pass < 2; ++pass) {           // 0: up, 1: down
        // ---- forward FFT of rectified signal (build fused into stage 0) ----
        k_fft_r8_fwd<<<gR8, blk, 0, stream>>>(x, Bb, pass == 0 ? 1.0f : -1.0f);
        {
            float2 *s = Bb, *d = A;
            int nsShift = 3;                          // Ns = 8 after radix-8
            for (int st = 0; st < 5; ++st) {          // Ns: 8,128,2048,32768,524288
                float inv16Ns = 1.0f / (16.0f * (float)(1 << nsShift));
                k_fft_r16_wmma<<<gR16, blk, 0, stream>>>(s, d, nsShift, -1.0f, inv16Ns);
                float2* t2 = s; s = d; d = t2;
                nsShift += 4;
            }                                         // spectrum lands in A
        }
        // ---- inverse FFT with Hilbert mask fused into stage 0 ----
        k_fft_r8_inv<<<gR8, blk, 0, stream>>>(A, Bb);
        {
            float2 *s = Bb, *d = A;
            int nsShift = 3;
            for (int st = 0; st < 5; ++st) {
                float inv16Ns = 1.0f / (16.0f * (float)(1 << nsShift));
                k_fft_r16_wmma<<<gR16, blk, 0, stream>>>(s, d, nsShift, 1.0f, inv16Ns);
                float2* t2 = s; s = d; d = t2;
                nsShift += 4;
            }                                         // analytic signal in A
        }
        k_env<<<gElem, blk, 0, stream>>>(A, out2, pass);
    }
    k_std<<<2048, blk, 0, stream>>>(x, out2, stds);
    k_final<<<gElem, blk, 0, stream>>>(x, out2, stds, out1);
}